// KNN_11141145166317
// MI455X (gfx1250) — compile-verified
//
#include <hip/hip_runtime.h>

// KNN top-k=20 smallest per row via exact 4x8-bit radix select.
// CDNA5 paths: global_load_async_to_lds_b128 (+ s_wait_asynccnt), LDS atomics,
// wave32 shuffle reductions. No WMMA: the op has no matrix-multiply structure.

#define NCOL 4096
#define KSEL 20
#define TPB  256
#define EPT  16     // NCOL / TPB elements per thread

__global__ __launch_bounds__(TPB)
void knn_topk_radix(const float* __restrict__ in, int* __restrict__ out) {
  __shared__ unsigned buf[NCOL];                 // async-DMA dest; later tie-index buffer
  __shared__ unsigned hist[256];
  __shared__ unsigned long long red8[TPB / 32];
  __shared__ unsigned long long winners[KSEL];
  __shared__ unsigned bc[2];                     // {bin, new_base} broadcast
  __shared__ unsigned cnt[2];                    // {nwin, ntie}

  const int tid = threadIdx.x;
  const unsigned row = blockIdx.x;
  const float* rowp = in + (size_t)row * NCOL;

  // ---- CDNA5 async DMA: 16 KB row -> LDS, 4 x b128 per thread, tracked by ASYNCcnt
  unsigned ldsBase = (unsigned)(size_t)&buf[0];
  #pragma unroll
  for (int j = 0; j < 4; ++j) {
    unsigned goff = (unsigned)(tid + j * TPB) * 16u;   // 16-byte chunks, fully coalesced
    unsigned loff = ldsBase + goff;
    asm volatile("global_load_async_to_lds_b128 %0, %1, %2 offset:0"
                 :: "v"(loff), "v"(goff), "s"(rowp) : "memory");
  }
  asm volatile("s_wait_asynccnt 0" ::: "memory");
  __syncthreads();

  // ---- order-preserving u32 keys into registers (strided: bank-conflict-free)
  unsigned vk[EPT];
  #pragma unroll
  for (int s = 0; s < EPT; ++s) {
    unsigned b = buf[tid + (s << 8)];
    vk[s] = b ^ ((unsigned)((int)b >> 31) | 0x80000000u);
  }
  __syncthreads();   // buf free for reuse after this

  // ---- 4-level radix descent: exact 20th-smallest key T, base = #keys < T
  unsigned prefix = 0, base = 0;

  // level 0 (top 8 bits)
  hist[tid] = 0u;
  __syncthreads();
  #pragma unroll
  for (int s = 0; s < EPT; ++s) atomicAdd(&hist[vk[s] >> 24], 1u);
  __syncthreads();
  if (tid == 0) {
    unsigned c = 0, b = 0;
    for (; b < 256u; ++b) { unsigned h = hist[b]; if (c + h >= (unsigned)KSEL) break; c += h; }
    bc[0] = b; bc[1] = c;
  }
  __syncthreads();
  prefix = bc[0] << 24; base = bc[1];
  __syncthreads();

  // levels 1..3
  #pragma unroll
  for (int lvl = 1; lvl < 4; ++lvl) {
    const int shift = 24 - 8 * lvl;                    // 16, 8, 0 (compile-time via unroll)
    hist[tid] = 0u;
    __syncthreads();
    #pragma unroll
    for (int s = 0; s < EPT; ++s) {
      unsigned k = vk[s];
      if ((k >> (shift + 8)) == (prefix >> (shift + 8)))
        atomicAdd(&hist[(k >> shift) & 255u], 1u);
    }
    __syncthreads();
    if (tid == 0) {
      unsigned c = base, b = 0;
      for (; b < 256u; ++b) { unsigned h = hist[b]; if (c + h >= (unsigned)KSEL) break; c += h; }
      bc[0] = b; bc[1] = c;
    }
    __syncthreads();
    prefix |= bc[0] << shift; base = bc[1];
    __syncthreads();
  }

  const unsigned T = prefix;
  const unsigned need = (unsigned)KSEL - base;   // >= 1 by construction

  // ---- filter pass: winners (<T) and tie candidates (==T)
  if (tid == 0) { cnt[0] = 0u; cnt[1] = 0u; }
  __syncthreads();
  #pragma unroll
  for (int s = 0; s < EPT; ++s) {
    unsigned k = vk[s];
    unsigned e = (unsigned)tid + ((unsigned)s << 8);
    if (k < T) {
      unsigned p = atomicAdd(&cnt[0], 1u);       // exactly `base` of these (<= 19)
      winners[p] = ((unsigned long long)k << 32) | e;
    } else if (k == T) {
      unsigned p = atomicAdd(&cnt[1], 1u);
      buf[p] = e;                                 // tie pool (usually size 1)
    }
  }
  __syncthreads();
  const unsigned ntie = cnt[1];

  // ---- extract `need` smallest-index ties (wave32 shuffle argmin; usually 1 iter)
  for (unsigned r = 0; r < need; ++r) {
    unsigned long long best = ~0ull;
    for (unsigned i = tid; i < ntie; i += TPB) {
      unsigned long long kk = ((unsigned long long)buf[i] << 32) | i;
      best = kk < best ? kk : best;
    }
    #pragma unroll
    for (int off = 16; off > 0; off >>= 1) {
      unsigned long long o = __shfl_xor(best, off, 32);
      best = o < best ? o : best;
    }
    if ((tid & 31) == 0) red8[tid >> 5] = best;
    __syncthreads();
    unsigned long long win = red8[0];
    #pragma unroll
    for (int w = 1; w < TPB / 32; ++w) { unsigned long long o = red8[w]; win = o < win ? o : win; }
    if (tid == 0) {
      winners[base + r] = ((unsigned long long)T << 32) | (win >> 32);
      buf[(unsigned)win] = 0xFFFFFFFFu;          // remove from candidate pool
    }
    __syncthreads();
  }

  // ---- rank-sort the 20 (key,index) pairs, emit indices ascending by (value,index)
  if (tid < KSEL) {
    unsigned long long x = winners[tid];
    int rank = 0;
    #pragma unroll
    for (int j = 0; j < KSEL; ++j) rank += (winners[j] < x) ? 1 : 0;
    out[(size_t)row * KSEL + rank] = (int)(x & 0xFFFFu);
  }
}

extern "C" void kernel_launch(void* const* d_in, const int* in_sizes, int n_in,
                              void* d_out, int out_size, void* d_ws, size_t ws_size,
                              hipStream_t stream) {
  const float* in = (const float*)d_in[0];
  int* out = (int*)d_out;
  const unsigned rows = (unsigned)((long long)in_sizes[0] / NCOL);   // 8*4096 = 32768
  hipLaunchKernelGGL(knn_topk_radix, dim3(rows), dim3(TPB), 0, stream, in, out);
}